// SpectralInterpolate_78185584656424
// MI455X (gfx1250) — compile-verified
//
#include <hip/hip_runtime.h>
#include <math.h>

typedef __attribute__((ext_vector_type(2))) float v2f;
typedef __attribute__((ext_vector_type(8))) float v8f;

static constexpr int Bn = 8, Cn = 64, Tn = 1200;
static constexpr int SEGS = 4;                 // K-segments for recon pass
static constexpr float TWO_PI_F = 6.28318530717958647692f;

// ---------------------------------------------------------------------------
// Kernel A: wtau[b,k] = 0.5 * atan2( sum_t m*sin(2 w t), sum_t m*cos(2 w t) )
// theta = omega*t - wtau.  One wave per (b,k).
// 2*omega_k*t = 2*pi*((k+1)*t mod 4800)/4800 exactly (k*t < 2^24).
// ---------------------------------------------------------------------------
__global__ void k_tau(const int* __restrict__ mask, float* __restrict__ wtau, int K) {
    int wave = (int)((blockIdx.x * blockDim.x + threadIdx.x) >> 5);
    int lane = threadIdx.x & 31;
    int total = Bn * K;
    if (wave >= total) return;
    int b = wave / K, k = wave % K;
    int kk = k + 1;
    const int* mrow = mask + b * Tn;
    const float sc = TWO_PI_F / 4800.0f;
    float s2 = 0.f, c2 = 0.f;
    for (int t = lane; t < Tn; t += 32) {
        float m = (float)mrow[t];
        int p = kk * (t + 1);
        float a = (float)(p % 4800) * sc;
        s2 += m * __sinf(a);
        c2 += m * __cosf(a);
    }
    #pragma unroll
    for (int off = 16; off > 0; off >>= 1) {
        s2 += __shfl_xor(s2, off, 32);
        c2 += __shfl_xor(c2, off, 32);
    }
    if (lane == 0) wtau[wave] = 0.5f * atan2f(s2, c2);
}

// ---------------------------------------------------------------------------
// Kernel B: per (b, 16-frequency tile):
//   num[k,c]  = sum_t basis[k,t] * dm[c,t]   via V_WMMA_F32_16X16X4_F32
//   den[k]    = sum_t basis[k,t]^2 * m[t]    via per-lane VALU + shfl fold
//   coef[k,c] = num/den  -> packed layout [b][k/4][c][k%4] (for b64 loads in C)
// A tile (16x4 f32): v0: lanes0-15 K=0 / lanes16-31 K=2 ; v1: K=1 / K=3.
// ---------------------------------------------------------------------------
__global__ void k_coef(const float* __restrict__ data, const int* __restrict__ mask,
                       const float* __restrict__ wtau,
                       float* __restrict__ cCoefP, float* __restrict__ sCoefP,
                       int K, int numKT, int K4cnt) {
    int wave = (int)((blockIdx.x * blockDim.x + threadIdx.x) >> 5);
    int lane = threadIdx.x & 31;
    int total = Bn * numKT;
    if (wave >= total) return;
    int b = wave / numKT, kt = wave % numKT;
    int half = lane >> 4;       // 0: lanes 0-15, 1: lanes 16-31
    int l16  = lane & 15;
    int krow = kt * 16 + l16;   // A-matrix row handled by this lane
    bool kvalid = krow < K;
    int kk = kvalid ? (krow + 1) : 1;
    float wt = kvalid ? wtau[(size_t)b * K + krow] : 0.f;

    v8f zero = {};
    v8f accC[4], accS[4];
    #pragma unroll
    for (int i = 0; i < 4; ++i) { accC[i] = zero; accS[i] = zero; }

    const float scA = TWO_PI_F / 9600.0f;
    const int* mrow = mask + b * Tn;
    float dC = 0.f, dS = 0.f;

    for (int t0 = 0; t0 < Tn; t0 += 4) {
        int tA = t0 + half * 2;            // lane's first 0-based timestep
        float m0 = (float)mrow[tA];
        float m1 = (float)mrow[tA + 1];

        int p0 = kk * (tA + 1);
        float th0 = (float)(p0 % 9600) * scA - wt;
        int p1 = kk * (tA + 2);
        float th1 = (float)(p1 % 9600) * scA - wt;
        float c0 = __cosf(th0), s0 = __sinf(th0);
        float c1 = __cosf(th1), s1 = __sinf(th1);
        if (!kvalid) { c0 = c1 = s0 = s1 = 0.f; }

        v2f aC = {c0, c1};
        v2f aS = {s0, s1};
        dC += m0 * c0 * c0 + m1 * c1 * c1;
        dS += m0 * s0 * s0 + m1 * s1 * s1;

        #pragma unroll
        for (int ct = 0; ct < 4; ++ct) {
            int c = ct * 16 + l16;
            const float* dp = data + ((size_t)(b * Cn + c)) * Tn;
            v2f bm;
            bm.x = dp[tA] * m0;
            bm.y = dp[tA + 1] * m1;
            accC[ct] = __builtin_amdgcn_wmma_f32_16x16x4_f32(
                false, aC, false, bm, (short)0, accC[ct], false, false);
            accS[ct] = __builtin_amdgcn_wmma_f32_16x16x4_f32(
                false, aS, false, bm, (short)0, accS[ct], false, false);
        }
    }

    // fold the two time-halves: lane L then holds full denom for row (L&15)
    float dCf = dC + __shfl_xor(dC, 16, 32);
    float dSf = dS + __shfl_xor(dS, 16, 32);

    // C tile: VGPR i -> element (M = i + half*8, N = l16)
    #pragma unroll
    for (int i = 0; i < 8; ++i) {
        int M = i + half * 8;
        int kr = kt * 16 + M;
        float denC = __shfl(dCf, M, 32);
        float denS = __shfl(dSf, M, 32);
        if (kr < K) {
            #pragma unroll
            for (int ct = 0; ct < 4; ++ct) {
                int c = ct * 16 + l16;
                size_t idx = (((size_t)b * K4cnt + (kr >> 2)) * Cn + c) * 4 + (kr & 3);
                cCoefP[idx] = accC[ct][i] / denC;
                sCoefP[idx] = accS[ct][i] / denS;
            }
        }
    }
}

// ---------------------------------------------------------------------------
// Kernel C: recon^T[t,c] = sum_k cos[k,t]*cC[k,c] + sin[k,t]*sC[k,c]
// One wave per (seg, b, 16-timestep tile); each segment reduces a disjoint
// K-range into its own partial buffer (deterministic, no atomics).
// Packed coef layout makes each B-tile fetch a single aligned b64 load.
// ---------------------------------------------------------------------------
__global__ void k_recon(const float* __restrict__ wtau,
                        const float* __restrict__ cCoefP, const float* __restrict__ sCoefP,
                        float* __restrict__ reconPart, int K, int K4cnt, int segLen) {
    const int numTT = Tn / 16; // 75
    int wave = (int)((blockIdx.x * blockDim.x + threadIdx.x) >> 5);
    int lane = threadIdx.x & 31;
    int total = SEGS * Bn * numTT;
    if (wave >= total) return;
    int seg = wave / (Bn * numTT);
    int rem = wave % (Bn * numTT);
    int b = rem / numTT, tt = rem % numTT;
    int kBeg = seg * segLen;
    int kEnd = min(K, kBeg + segLen);
    int kEnd4 = kBeg + ((kEnd - kBeg) & ~3);   // guard-free main range

    int half = lane >> 4, l16 = lane & 15;
    int trow = tt * 16 + l16;       // A row = 0-based timestep
    int tval = trow + 1;
    const float scA = TWO_PI_F / 9600.0f;
    const float* wt_b = wtau + (size_t)b * K;
    size_t coefBase = (size_t)b * K4cnt * Cn * 4;

    v8f zero = {};
    v8f acc[4];
    #pragma unroll
    for (int i = 0; i < 4; ++i) acc[i] = zero;

    // main loop: no bounds checks, b64 coefficient loads
    for (int kc = kBeg; kc < kEnd4; kc += 4) {
        int kA0 = kc + half * 2;    // this lane's two K-columns: kA0, kA0+1
        float wt0 = wt_b[kA0];
        float wt1 = wt_b[kA0 + 1];
        int p0 = (kA0 + 1) * tval;
        float th0 = (float)(p0 % 9600) * scA - wt0;
        int p1 = (kA0 + 2) * tval;
        float th1 = (float)(p1 % 9600) * scA - wt1;
        v2f aC = {__cosf(th0), __cosf(th1)};
        v2f aS = {__sinf(th0), __sinf(th1)};

        size_t rowBase = coefBase + (size_t)(kc >> 2) * Cn * 4;
        #pragma unroll
        for (int ct = 0; ct < 4; ++ct) {
            int c = ct * 16 + l16;
            size_t i0 = rowBase + (size_t)c * 4 + half * 2;   // 8B aligned
            v2f bC = *(const v2f*)(cCoefP + i0);
            v2f bS = *(const v2f*)(sCoefP + i0);
            acc[ct] = __builtin_amdgcn_wmma_f32_16x16x4_f32(
                false, aC, false, bC, (short)0, acc[ct], false, false);
            acc[ct] = __builtin_amdgcn_wmma_f32_16x16x4_f32(
                false, aS, false, bS, (short)0, acc[ct], false, false);
        }
    }

    // tail (at most one partially-valid chunk)
    for (int kc = kEnd4; kc < kEnd; kc += 4) {
        int kA0 = kc + half * 2;
        bool v0 = kA0 < kEnd, v1 = (kA0 + 1) < kEnd;
        float wt0 = v0 ? wt_b[kA0] : 0.f;
        float wt1 = v1 ? wt_b[kA0 + 1] : 0.f;
        int p0 = (kA0 + 1) * tval;
        float th0 = (float)(p0 % 9600) * scA - wt0;
        int p1 = (kA0 + 2) * tval;
        float th1 = (float)(p1 % 9600) * scA - wt1;
        float c0 = v0 ? __cosf(th0) : 0.f;
        float s0 = v0 ? __sinf(th0) : 0.f;
        float c1 = v1 ? __cosf(th1) : 0.f;
        float s1 = v1 ? __sinf(th1) : 0.f;
        v2f aC = {c0, c1};
        v2f aS = {s0, s1};
        size_t rowBase = coefBase + (size_t)(kc >> 2) * Cn * 4;
        #pragma unroll
        for (int ct = 0; ct < 4; ++ct) {
            int c = ct * 16 + l16;
            size_t i0 = rowBase + (size_t)c * 4 + half * 2;
            v2f bC, bS;
            bC.x = v0 ? cCoefP[i0] : 0.f;
            bC.y = v1 ? cCoefP[i0 + 1] : 0.f;
            bS.x = v0 ? sCoefP[i0] : 0.f;
            bS.y = v1 ? sCoefP[i0 + 1] : 0.f;
            acc[ct] = __builtin_amdgcn_wmma_f32_16x16x4_f32(
                false, aC, false, bC, (short)0, acc[ct], false, false);
            acc[ct] = __builtin_amdgcn_wmma_f32_16x16x4_f32(
                false, aS, false, bS, (short)0, acc[ct], false, false);
        }
    }

    float* rseg = reconPart + (size_t)seg * Bn * Cn * Tn;
    #pragma unroll
    for (int i = 0; i < 8; ++i) {
        int t = tt * 16 + i + half * 8;
        #pragma unroll
        for (int ct = 0; ct < 4; ++ct) {
            int c = ct * 16 + l16;
            rseg[((size_t)(b * Cn + c)) * Tn + t] = acc[ct][i];
        }
    }
}

// ---------------------------------------------------------------------------
// Kernel D: sum recon partials, normalize to observed std, masked merge.
// var_orig = (sum m d^2 - cnt*mu^2)/(cnt-1), mu = sum(m d)/cnt
// var_rec  = (sum r^2 - T*mr^2)/(T-1)
// ---------------------------------------------------------------------------
__global__ void k_final(const float* __restrict__ data, const int* __restrict__ mask,
                        const float* __restrict__ reconPart, float* __restrict__ out) {
    const size_t PS = (size_t)Bn * Cn * Tn;     // partial stride
    int bc = blockIdx.x;            // 0 .. B*C-1
    int b = bc / Cn;
    const float* dp = data + (size_t)bc * Tn;
    const float* rp = reconPart + (size_t)bc * Tn;
    const int* mp = mask + b * Tn;

    float sr = 0.f, srr = 0.f, smd = 0.f, smdd = 0.f, sm = 0.f;
    for (int t = threadIdx.x; t < Tn; t += blockDim.x) {
        float r = rp[t] + rp[t + PS] + rp[t + 2 * PS] + rp[t + 3 * PS];
        float d = dp[t];
        float m = (float)mp[t];
        sr += r; srr += r * r;
        smd += m * d; smdd += m * d * d; sm += m;
    }
    #pragma unroll
    for (int off = 16; off > 0; off >>= 1) {
        sr += __shfl_xor(sr, off, 32);
        srr += __shfl_xor(srr, off, 32);
        smd += __shfl_xor(smd, off, 32);
        smdd += __shfl_xor(smdd, off, 32);
        sm += __shfl_xor(sm, off, 32);
    }
    __shared__ float g0[8], g1[8], g2[8], g3[8], g4[8];
    __shared__ float ratio_s;
    int wid = threadIdx.x >> 5, ln = threadIdx.x & 31;
    if (ln == 0) { g0[wid] = sr; g1[wid] = srr; g2[wid] = smd; g3[wid] = smdd; g4[wid] = sm; }
    __syncthreads();
    if (threadIdx.x == 0) {
        float tsr = 0.f, tsrr = 0.f, tsmd = 0.f, tsmdd = 0.f, tsm = 0.f;
        int nw = (blockDim.x + 31) >> 5;
        for (int i = 0; i < nw; ++i) { tsr += g0[i]; tsrr += g1[i]; tsmd += g2[i]; tsmdd += g3[i]; tsm += g4[i]; }
        float cnt = tsm;
        float mu = tsmd / cnt;
        float var_o = (tsmdd - cnt * mu * mu) / (cnt - 1.f);
        float mr = tsr / (float)Tn;
        float var_r = (tsrr - (float)Tn * mr * mr) / ((float)Tn - 1.f);
        ratio_s = sqrtf(var_o) / sqrtf(var_r);
    }
    __syncthreads();
    float ratio = ratio_s;
    for (int t = threadIdx.x; t < Tn; t += blockDim.x) {
        float r = rp[t] + rp[t + PS] + rp[t + 2 * PS] + rp[t + 3 * PS];
        out[(size_t)bc * Tn + t] = mp[t] ? dp[t] : r * ratio;
    }
}

// ---------------------------------------------------------------------------
extern "C" void kernel_launch(void* const* d_in, const int* in_sizes, int n_in,
                              void* d_out, int out_size, void* d_ws, size_t ws_size,
                              hipStream_t stream) {
    const float* data = (const float*)d_in[0];
    const int* mask = (const int*)d_in[1];
    float* out = (float*)d_out;

    // K replicates np.arange(step, 0.5+step, step) length in float64.
    const double step = 1.0 / (1200.0 * 8.0);     // 1/(timespan*OSFREQ)
    const double start = step;
    const double stop = 1.0 * 1200.0 / (2.0 * 1200.0) + step;  // 0.5 + step
    int K = (int)ceil((stop - start) / step);
    if (K < 1) K = 1;
    int K4cnt = (K + 3) / 4;
    int numKT = (K + 15) / 16;
    int segLen = (((K + SEGS - 1) / SEGS) + 3) & ~3;  // multiple of 4

    float* ws = (float*)d_ws;
    float* wtau   = ws;                                      // B*K
    float* cCoefP = wtau + (size_t)Bn * K;                   // B*K4cnt*C*4
    float* sCoefP = cCoefP + (size_t)Bn * K4cnt * Cn * 4;    // B*K4cnt*C*4
    float* reconP = sCoefP + (size_t)Bn * K4cnt * Cn * 4;    // SEGS*B*C*T

    {   // Pass A: omega*tau per (b,k)
        int waves = Bn * K;
        int blocks = (waves * 32 + 255) / 256;
        k_tau<<<blocks, 256, 0, stream>>>(mask, wtau, K);
    }
    {   // Pass B: coefficients via WMMA (packed layout)
        int waves = Bn * numKT;
        int blocks = (waves * 32 + 255) / 256;
        k_coef<<<blocks, 256, 0, stream>>>(data, mask, wtau, cCoefP, sCoefP, K, numKT, K4cnt);
    }
    {   // Pass C: reconstruction via WMMA, K split into SEGS partials
        int waves = SEGS * Bn * (Tn / 16);
        int blocks = (waves * 32 + 255) / 256;
        k_recon<<<blocks, 256, 0, stream>>>(wtau, cCoefP, sCoefP, reconP, K, K4cnt, segLen);
    }
    // Pass D: partial-sum + normalization + masked merge
    k_final<<<Bn * Cn, 256, 0, stream>>>(data, mask, reconP, out);
}